// TemporalEncoding_21749714387279
// MI455X (gfx1250) — compile-verified
//
#include <hip/hip_runtime.h>
#include <stdint.h>

// TemporalEncoding broadcast kernel for MI455X (gfx1250).
//
// Reference: out[b, r, e] = temp_emb[r, e] for all b in [0,128).
//   temp_emb row map:
//     r == 0          -> zeros              (modal0 "Global" row)
//     1   <= r <= 256 -> pe[r-1]            (linspace(0,255,256) == identity)
//     r == 257        -> zeros              (modal1 "Global" row)
//     258 <= r <= 385 -> pe[trunc(j*255/127)], j = r-258, endpoint j==127 -> 255
//
// Pure store-bandwidth problem: ~101 MB written, ~0.8 MB read (pe is L2-resident).
// Strategy: one block per (row, 16-batch chunk); stage the 2KB pe row into LDS
// via gfx1250 async global->LDS b128 copies (ASYNCcnt), then blast 8x b128
// non-temporal stores per thread.

typedef float v4f __attribute__((ext_vector_type(4)));

#define D_MODEL       512
#define N_ROWS        386          // (256+1) + (128+1)
#define N_BATCH       128
#define F4_PER_ROW    (D_MODEL / 4)   // 128 float4 slots per row
#define BATCH_PER_BLK 16

__global__ __launch_bounds__(256)
void temporal_pe_bcast_kernel(const float* __restrict__ pe,
                              float* __restrict__ out) {
    const int r      = blockIdx.x;              // output row, 0..385
    const int bchunk = blockIdx.y;              // 0..7 (16 batches each)
    const int tid    = threadIdx.x;             // 0..255 (8 wave32s)
    const int col4   = tid & (F4_PER_ROW - 1);  // float4 column slot 0..127
    const int bsub   = tid >> 7;                // 0 or 1

    __shared__ v4f srow[F4_PER_ROW];            // one staged 512-float row (2KB)

    // ---- map output row -> pe source row (or -1 for a zero row) ----
    int src;
    if (r == 0 || r == 257) {
        src = -1;
    } else if (r <= 256) {
        src = r - 1;                            // exact-integer linspace
    } else {
        const int j = r - 258;                  // 0..127
        // np.linspace(0,255,128): y[j] = j*(255/127) in f64, truncated to i32,
        // with the endpoint pinned to exactly 255.
        src = (j == 127) ? 255 : (int)((double)j * (255.0 / 127.0));
    }

    // ---- stage the row into LDS (gfx1250 async copy path) ----
    if (tid < F4_PER_ROW) {
        // Low 32 bits of a generic pointer to LDS are the LDS byte offset
        // (CDNA5 ISA 10.2).
        const uint32_t lds_addr = (uint32_t)(uintptr_t)(&srow[col4]);
        if (src >= 0) {
            const uint64_t gaddr =
                (uint64_t)(uintptr_t)(pe + (size_t)src * D_MODEL + (size_t)col4 * 4);
            // 16 bytes per lane, 4 waves -> whole 2KB row; tracked by ASYNCcnt.
            asm volatile("global_load_async_to_lds_b128 %0, %1, off"
                         :
                         : "v"(lds_addr), "v"(gaddr)
                         : "memory");
            asm volatile("s_wait_asynccnt 0" ::: "memory");
        } else {
            v4f z = {0.0f, 0.0f, 0.0f, 0.0f};
            srow[col4] = z;
        }
    }
    __syncthreads();

    const v4f v = srow[col4];

    // ---- broadcast: 8 non-temporal b128 stores per thread ----
    const size_t bstride = (size_t)N_ROWS * (size_t)D_MODEL;   // 386*512 floats
    float* const base = out + (size_t)r * D_MODEL + (size_t)col4 * 4;
    const int b0 = bchunk * BATCH_PER_BLK + bsub;
#pragma unroll
    for (int i = 0; i < BATCH_PER_BLK / 2; ++i) {
        const int b = b0 + i * 2;
        __builtin_nontemporal_store(v, (v4f*)(base + (size_t)b * bstride));
    }
}

extern "C" void kernel_launch(void* const* d_in, const int* in_sizes, int n_in,
                              void* d_out, int out_size, void* d_ws, size_t ws_size,
                              hipStream_t stream) {
    // setup_inputs order: modal0 (unused, shape-only), modal1 (unused, shape-only), pe
    const float* pe  = (const float*)d_in[2];
    float*       out = (float*)d_out;
    (void)in_sizes; (void)n_in; (void)out_size; (void)d_ws; (void)ws_size;

    dim3 grid(N_ROWS, N_BATCH / BATCH_PER_BLK);   // (386, 8)
    dim3 block(256);
    temporal_pe_bcast_kernel<<<grid, block, 0, stream>>>(pe, out);
}